// variableAttention_89644557403043
// MI455X (gfx1250) — compile-verified
//
#include <hip/hip_runtime.h>
#include <math.h>
#include <stdint.h>

// ---- Problem constants (from reference) ----
#define B_ 32
#define F_ 32
#define T_ 2048
#define D_ 64
#define K_ 3

#define NTHREADS 256          // 8 wave32s per block
#define NWAVES   8
#define TCHUNK   128          // t per block iteration = NWAVES * 16
#define NCHUNK   (T_ / TCHUNK) // 16
#define SVSTRIDE 65           // pad D=64 rows to dodge 64-bank LDS conflicts
#define XPAD     2            // zero padding on both ends of x row in LDS

typedef __attribute__((ext_vector_type(2))) float v2f;
typedef __attribute__((ext_vector_type(8))) float v8f;

__device__ __forceinline__ float sigmoidf_(float v) {
    return 1.0f / (1.0f + __expf(-v));
}

// One block per (b,f). Fused: conv (via v_wmma_f32_16x16x4_f32) -> Tatt ->
// res -> vAtt -> writes atten (scaled) + res/vAtt workspace.
__global__ __launch_bounds__(NTHREADS) void va_main_kernel(
    const float* __restrict__ x,        // [B,F,T]
    const float* __restrict__ conv_w,   // [F,D,K]
    const float* __restrict__ conv_b,   // [F,D]
    const float* __restrict__ W1,       // [F,D]
    const float* __restrict__ b1,       // [F,T]
    const float* __restrict__ W2,       // [D]
    const float* __restrict__ b2,       // [F]
    float* __restrict__ atten,          // [B,F,T] (already scaled by vAtt)
    float* __restrict__ res_ws,         // [B*F, D]
    float* __restrict__ vatt_ws)        // [B*F]
{
    __shared__ float x_lds[XPAD + T_ + XPAD];      // zero-padded x row (8.02 KB)
    __shared__ float sv_lds[TCHUNK * SVSTRIDE];    // 33.3 KB: one 128x64 sv chunk
    __shared__ float w1_lds[D_];
    __shared__ float tatt_lds[TCHUNK];
    __shared__ float red_lds[NTHREADS];
    __shared__ float vatt_sh;

    const int tid  = threadIdx.x;
    const int bf   = blockIdx.x;          // b*F + f
    const int f    = bf & (F_ - 1);
    const int wave = tid >> 5;
    const int lane = tid & 31;
    const int lanem = lane & 15;
    const bool hi  = lane >= 16;

    // ---- Stage x[b,f,:] into LDS via CDNA5 async global->LDS path ----
    // GLOBAL_LOAD_ASYNC_TO_LDS_B32: VDST = per-lane LDS byte address,
    // VADDR = 64-bit global address; tracked with ASYNCcnt (no VGPR bounce).
    const float* xrow = x + (size_t)bf * T_;
    for (int i = tid; i < T_; i += NTHREADS) {
        unsigned ldsb = (unsigned)(uintptr_t)(&x_lds[XPAD + i]);
        const float* gp = xrow + i;
        asm volatile("global_load_async_to_lds_b32 %0, %1, off"
                     :: "v"(ldsb), "v"(gp) : "memory");
    }
    // zero halo: indices [0,2) and [XPAD+T, XPAD+T+2)
    if (tid < XPAD) {
        x_lds[tid] = 0.0f;
        x_lds[XPAD + T_ + tid] = 0.0f;
    }
    if (tid < D_) w1_lds[tid] = W1[f * D_ + tid];
    asm volatile("s_wait_asynccnt 0x0" ::: "memory");  // drain this wave's async loads
    __syncthreads();

    // ---- Build B fragments (conv weights) per ISA f32 4x16 B layout ----
    // lanes 0-15: (K=0,K=1) for N=lane; lanes 16-31: (K=2, K=3==0) for N=lane-16
    v2f  bfrag[4];
    float biasv[4];
#pragma unroll
    for (int n = 0; n < 4; ++n) {
        const int d = n * 16 + lanem;
        const float* wd = conv_w + (size_t)(f * D_ + d) * K_;
        if (!hi) { bfrag[n].x = wd[0]; bfrag[n].y = wd[1]; }
        else     { bfrag[n].x = wd[2]; bfrag[n].y = 0.0f; }   // K=3 pad row = 0
        biasv[n] = conv_b[f * D_ + d];
    }

    float tatt_reg[NCHUNK];   // only threads < TCHUNK use these
    float racc = 0.0f;        // partial res accumulator
    const int dmine = tid & (D_ - 1);
    const int tgrp  = tid >> 6;   // 4 groups of 32 t's each

    for (int c = 0; c < NCHUNK; ++c) {
        const int t0 = c * TCHUNK;
        const int tb = t0 + wave * 16;      // this wave's 16 t-rows

        // ---- A fragment: A[m,k] = x[tb + m - 2 + k]; padded index = tb+m+k.
        // lanes 0-15 hold K=0,1; lanes 16-31 hold K=2,3 (row m = lanem).
        // Padding makes every index in range -> two unconditional LDS loads.
        const int i0 = tb + lanem + (hi ? 2 : 0);
        v2f a;
        a.x = x_lds[i0];
        a.y = x_lds[i0 + 1];

        // ---- Conv as 4x v_wmma_f32_16x16x4_f32, bias preloaded in C ----
#pragma unroll
        for (int n = 0; n < 4; ++n) {
            v8f cacc;
#pragma unroll
            for (int r = 0; r < 8; ++r) cacc[r] = biasv[n];
            cacc = __builtin_amdgcn_wmma_f32_16x16x4_f32(
                false, a, false, bfrag[n], (short)0, cacc, false, false);
            const int d = n * 16 + lanem;
#pragma unroll
            for (int r = 0; r < 8; ++r) {
                const int mrow = hi ? (r + 8) : r;  // C VGPR r -> rows r / r+8
                sv_lds[(wave * 16 + mrow) * SVSTRIDE + d] = cacc[r];
            }
        }
        __syncthreads();

        // ---- Tatt logits: one thread per t in chunk ----
        if (tid < TCHUNK) {
            float acc = 0.0f;
#pragma unroll 8
            for (int d = 0; d < D_; ++d)
                acc += sv_lds[tid * SVSTRIDE + d] * w1_lds[d];
            const float tt = sigmoidf_(acc + b1[f * T_ + t0 + tid]);
            tatt_reg[c]   = tt;
            tatt_lds[tid] = tt;
        }
        __syncthreads();

        // ---- res partial: thread owns d=dmine, t-group tgrp (32 t's) ----
#pragma unroll 4
        for (int j = 0; j < 32; ++j) {
            const int tl = tgrp * 32 + j;
            racc += tatt_lds[tl] * sv_lds[tl * SVSTRIDE + dmine];
        }
        __syncthreads();   // protect sv_lds/tatt_lds before next chunk's writes
    }

    // ---- Reduce res across the 4 t-groups ----
    red_lds[tid] = racc;
    __syncthreads();
    float resv = 0.0f;
    if (tid < D_) {
        resv = red_lds[tid] + red_lds[D_ + tid] +
               red_lds[2 * D_ + tid] + red_lds[3 * D_ + tid];
        red_lds[tid] = resv * W2[tid];   // safe: only [0,64) rewritten
    }
    __syncthreads();
    if (tid == 0) {
        float s = 0.0f;
        for (int d = 0; d < D_; ++d) s += red_lds[d];
        vatt_sh = sigmoidf_(s + b2[f]);
    }
    __syncthreads();
    const float vatt = vatt_sh;

    if (tid < D_) res_ws[(size_t)bf * D_ + tid] = resv;
    if (tid == 0) vatt_ws[bf] = vatt;

    // ---- atten = Tatt * vAtt, straight from registers (single write) ----
    if (tid < TCHUNK) {
        float* arow = atten + (size_t)bf * T_;
#pragma unroll
        for (int c = 0; c < NCHUNK; ++c)
            arow[c * TCHUNK + tid] = tatt_reg[c] * vatt;
    }
}

// Deterministic cross-f reduction: out[b,d] = sum_f vAtt[b,f]*res[b,f,d]
__global__ __launch_bounds__(D_) void va_out_kernel(
    const float* __restrict__ res_ws,   // [B*F, D]
    const float* __restrict__ vatt_ws,  // [B*F]
    float* __restrict__ out)            // [B, D]
{
    const int b = blockIdx.x;
    const int d = threadIdx.x;
    float acc = 0.0f;
    for (int f = 0; f < F_; ++f)
        acc += vatt_ws[b * F_ + f] * res_ws[(size_t)(b * F_ + f) * D_ + d];
    out[b * D_ + d] = acc;
}

extern "C" void kernel_launch(void* const* d_in, const int* in_sizes, int n_in,
                              void* d_out, int out_size, void* d_ws, size_t ws_size,
                              hipStream_t stream) {
    const float* x      = (const float*)d_in[0];
    const float* conv_w = (const float*)d_in[1];
    const float* conv_b = (const float*)d_in[2];
    const float* W1     = (const float*)d_in[3];
    const float* b1     = (const float*)d_in[4];
    const float* W2     = (const float*)d_in[5];
    const float* b2     = (const float*)d_in[6];

    float* out_p   = (float*)d_out;                 // [B,D] = 2048 floats
    float* atten_p = (float*)d_out + B_ * D_;       // [B,F,T]

    float* res_ws  = (float*)d_ws;                  // B*F*D floats
    float* vatt_ws = res_ws + (size_t)B_ * F_ * D_; // B*F floats

    va_main_kernel<<<B_ * F_, NTHREADS, 0, stream>>>(
        x, conv_w, conv_b, W1, b1, W2, b2, atten_p, res_ws, vatt_ws);
    va_out_kernel<<<B_, D_, 0, stream>>>(res_ws, vatt_ws, out_p);
}